// NodeEncoder_68693706932385
// MI455X (gfx1250) — compile-verified
//
#include <hip/hip_runtime.h>

// ---------------------------------------------------------------------------
// GraphSAGE (3 layers) for MI455X / gfx1250, wave32.
//   h1 = relu(mean(x)  @Wl1 + bl1 + x @Wr1)   21(pad 24) -> 256
//   h2 = relu(mean(h1) @Wl2 + bl2 + h1@Wr2)   256        -> 128
//   out=      mean(h2) @Wl3 + bl3 + h2@Wr3    128        -> 128
// Edge aggregation is the HBM/atomic-bound part; dense layers use
// V_WMMA_F32_16X16X4_F32 (full fp32) with weight panels staged in LDS.
// ---------------------------------------------------------------------------

typedef __attribute__((ext_vector_type(2))) float v2f;
typedef __attribute__((ext_vector_type(8))) float v8f;

#define N_NODES 50000
#define N_EDGES 800000
#define D0P 24              // padded layer-1 input dim (21 -> 24)

// ---------------------------------------------------------------- fill zero
__global__ void fill_zero4_kernel(float4* __restrict__ p, long long n4) {
    long long stride = (long long)gridDim.x * blockDim.x;
    const float4 z = make_float4(0.f, 0.f, 0.f, 0.f);
    for (long long i = (long long)blockIdx.x * blockDim.x + threadIdx.x; i < n4; i += stride)
        p[i] = z;
}

// ------------------------------------------------------------- pad helpers
__global__ void pad_x_kernel(const float* __restrict__ x, float* __restrict__ xp) {
    long long total  = (long long)N_NODES * D0P;
    long long stride = (long long)gridDim.x * blockDim.x;
    for (long long i = (long long)blockIdx.x * blockDim.x + threadIdx.x; i < total; i += stride) {
        int n = (int)(i / D0P), f = (int)(i - (long long)n * D0P);
        xp[i] = (f < 21) ? x[(long long)n * 21 + f] : 0.0f;
    }
}
__global__ void pad_w_kernel(const float* __restrict__ w, float* __restrict__ wp) {
    int i = blockIdx.x * blockDim.x + threadIdx.x;     // 24*256 = 6144
    if (i < D0P * 256) {
        int k = i >> 8;
        wp[i] = (k < 21) ? w[i] : 0.0f;
    }
}

// ---------------------------------------------------------------- degrees
__global__ void degree_kernel(const int* __restrict__ dst, float* __restrict__ deg,
                              int n_edges) {
    int stride = gridDim.x * blockDim.x;
    for (int e = blockIdx.x * blockDim.x + threadIdx.x; e < n_edges; e += stride)
        atomicAdd(&deg[dst[e]], 1.0f);
}

__global__ void invdeg_kernel(const float* __restrict__ deg, float* __restrict__ inv,
                              int n) {
    int i = blockIdx.x * blockDim.x + threadIdx.x;
    if (i < n) {
        float d = deg[i];
        inv[i] = (d > 0.0f) ? (1.0f / fmaxf(d, 1.0f)) : 0.0f;
    }
}

// ------------------------------------------------- edge scatter-sum (float4)
__global__ void scatter_add4_kernel(const float* __restrict__ feat,
                                    const int* __restrict__ src,
                                    const int* __restrict__ dst,
                                    float* __restrict__ agg,
                                    int n_edges, int dq) {   // dq = d/4
    long long total  = (long long)n_edges * dq;
    long long stride = (long long)gridDim.x * blockDim.x;
    for (long long i = (long long)blockIdx.x * blockDim.x + threadIdx.x; i < total; i += stride) {
        int e = (int)(i / dq);
        int q = (int)(i - (long long)e * dq);
        const float4 v = ((const float4*)(feat + (long long)src[e] * dq * 4))[q];
        float* base = agg + (long long)dst[e] * dq * 4 + q * 4;
        atomicAdd(base + 0, v.x);
        atomicAdd(base + 1, v.y);
        atomicAdd(base + 2, v.z);
        atomicAdd(base + 3, v.w);
    }
}

// ------------------------------------------------------- fused SAGE GEMM
//   OUT = inv_deg * (agg @ Wl) + h @ Wr + bias   (+ optional ReLU)
// Block = 8 waves, all sharing one 16-column tile tn over 8 consecutive row
// tiles tm. The two DINx16 weight panels are staged into LDS once per block;
// the K loop is then: A fragments via global_load_b64 (immediate offsets,
// DIN is a template constant), B fragments via bank-conflict-free ds_load,
// two interleaved V_WMMA_F32_16X16X4_F32 accumulator chains.
// Per-row mean scale folded into the epilogue ((s*A)@B == s*(A@B) row-wise).
//
// f32 WMMA VGPR layouts (ISA 7.12.2):
//   A (16x4, MxK): lane m / m+16 holds M=m; lanes 0-15 -> K={0,1}, 16-31 -> K={2,3}
//   B (4x16, KxN): lane n / n+16 holds N=n; lanes 0-15 -> K={0,1}, 16-31 -> K={2,3}
//   C/D (16x16):   VGPR r -> row (r + 8*half), col = lane&15
template <int DIN, int DOUT, bool RELU>
__global__ void __launch_bounds__(256)
sage_gemm_wmma_kernel(const float* __restrict__ agg,
                      const float* __restrict__ inv_deg,
                      const float* __restrict__ h,
                      const float* __restrict__ Wl,
                      const float* __restrict__ Wr,
                      const float* __restrict__ bias,
                      float* __restrict__ out) {
    constexpr int TILES_N = DOUT / 16;
    constexpr int TILES_M = N_NODES / 16;          // 3125 exactly
    __shared__ float sWl[DIN * 16];                // 16-column panel of Wl
    __shared__ float sWr[DIN * 16];                // 16-column panel of Wr

    const int tn   = blockIdx.x % TILES_N;
    const int tm0  = (blockIdx.x / TILES_N) * 8;   // 8 row tiles per block
    const int col0 = tn << 4;

    // ---- cooperative panel stage: sW[k][c] = W[k][col0+c] -----------------
    for (int i = threadIdx.x; i < DIN * 16; i += 256) {
        const int k = i >> 4, c = i & 15;
        sWl[i] = Wl[(long long)k * DOUT + col0 + c];
        sWr[i] = Wr[(long long)k * DOUT + col0 + c];
    }
    __syncthreads();                               // all waves reach this

    const int lane = threadIdx.x & 31;
    const int tm   = tm0 + (threadIdx.x >> 5);     // wave-uniform
    if (tm >= TILES_M) return;                     // EXEC stays all-1s

    const int row0 = tm << 4;
    const int half = lane >> 4;                    // 0: K={0,1}   1: K={2,3}
    const int m    = lane & 15;                    // A row / B,C,D col in tile

    v8f acc0 = {};                                 // agg @ Wl
    v8f acc1 = {};                                 // h   @ Wr

    const float* __restrict__ Arow0 = agg + (long long)(row0 + m) * DIN + 2 * half;
    const float* __restrict__ Arow1 = h   + (long long)(row0 + m) * DIN + 2 * half;
    const float* Bl = sWl + 2 * half * 16 + m;     // LDS, +k0*16 are immediates
    const float* Br = sWr + 2 * half * 16 + m;

#pragma unroll 8
    for (int k0 = 0; k0 < DIN; k0 += 4) {
        v2f a0 = *(const v2f*)(Arow0 + k0);        // global_load_b64, imm offset
        v2f a1 = *(const v2f*)(Arow1 + k0);
        v2f b0, b1;
        b0.x = Bl[k0 * 16];                        // ds_load, imm offsets
        b0.y = Bl[k0 * 16 + 16];
        b1.x = Br[k0 * 16];
        b1.y = Br[k0 * 16 + 16];
        acc0 = __builtin_amdgcn_wmma_f32_16x16x4_f32(false, a0, false, b0,
                                                     (short)0, acc0, false, false);
        acc1 = __builtin_amdgcn_wmma_f32_16x16x4_f32(false, a1, false, b1,
                                                     (short)0, acc1, false, false);
    }

    const float bv = bias[col0 + m];
#pragma unroll
    for (int r = 0; r < 8; ++r) {
        const int row = row0 + r + 8 * half;
        float v = acc0[r] * inv_deg[row] + acc1[r] + bv;
        if (RELU) v = fmaxf(v, 0.0f);
        out[(long long)row * DOUT + col0 + m] = v;
    }
}

// ---------------------------------------------------------------------------
extern "C" void kernel_launch(void* const* d_in, const int* in_sizes, int n_in,
                              void* d_out, int out_size, void* d_ws, size_t ws_size,
                              hipStream_t stream) {
    // inputs (setup_inputs order)
    const float* x    = (const float*)d_in[0];              // [50000, 21]
    const int*   ei   = (const int*)d_in[1];                // [2, 800000]
    const float* Wl1  = (const float*)d_in[2];              // [21, 256]
    const float* bl1  = (const float*)d_in[3];              // [256]
    const float* Wr1  = (const float*)d_in[4];              // [21, 256]
    const float* Wl2  = (const float*)d_in[5];              // [256, 128]
    const float* bl2  = (const float*)d_in[6];              // [128]
    const float* Wr2  = (const float*)d_in[7];              // [256, 128]
    const float* Wl3  = (const float*)d_in[8];              // [128, 128]
    const float* bl3  = (const float*)d_in[9];              // [128]
    const float* Wr3  = (const float*)d_in[10];             // [128, 128]
    const int* src = ei;                                    // row 0
    const int* dst = ei + N_EDGES;                          // row 1

    // workspace layout (floats); all sub-buffer sizes are multiples of 4
    float* ws      = (float*)d_ws;
    float* deg     = ws;                                    // [N]
    float* inv_deg = deg + N_NODES;                         // [N]
    float* xp      = inv_deg + N_NODES;                     // [N, 24]  padded x
    float* Wl1p    = xp + (long long)N_NODES * D0P;         // [24, 256]
    float* Wr1p    = Wl1p + D0P * 256;                      // [24, 256]
    float* agg     = Wr1p + D0P * 256;                      // [N, 256] max
    float* h1      = agg + (long long)N_NODES * 256;        // [N, 256]
    float* h2      = h1  + (long long)N_NODES * 256;        // [N, 128]
    float* outf    = (float*)d_out;                         // [N, 128]

    const int T = 256;
    const int ZB = 2048, SB = 8192;                         // grid-stride blocks
    const int MB8 = (N_NODES / 16 + 7) / 8;                 // row-tile groups: 391

    // --- padding + degrees -------------------------------------------------
    pad_x_kernel<<<ZB, T, 0, stream>>>(x, xp);
    pad_w_kernel<<<(D0P * 256 + T - 1) / T, T, 0, stream>>>(Wl1, Wl1p);
    pad_w_kernel<<<(D0P * 256 + T - 1) / T, T, 0, stream>>>(Wr1, Wr1p);
    fill_zero4_kernel<<<64, T, 0, stream>>>((float4*)deg, N_NODES / 4);
    degree_kernel<<<(N_EDGES + T - 1) / T, T, 0, stream>>>(dst, deg, N_EDGES);
    invdeg_kernel<<<(N_NODES + T - 1) / T, T, 0, stream>>>(deg, inv_deg, N_NODES);

    // --- layer 1: 24 -> 256, relu -----------------------------------------
    fill_zero4_kernel<<<ZB, T, 0, stream>>>((float4*)agg, (long long)N_NODES * D0P / 4);
    scatter_add4_kernel<<<SB, T, 0, stream>>>(xp, src, dst, agg, N_EDGES, D0P / 4);
    sage_gemm_wmma_kernel<D0P, 256, true>
        <<<MB8 * (256 / 16), T, 0, stream>>>(agg, inv_deg, xp, Wl1p, Wr1p, bl1, h1);

    // --- layer 2: 256 -> 128, relu ----------------------------------------
    fill_zero4_kernel<<<ZB, T, 0, stream>>>((float4*)agg, (long long)N_NODES * 256 / 4);
    scatter_add4_kernel<<<SB, T, 0, stream>>>(h1, src, dst, agg, N_EDGES, 256 / 4);
    sage_gemm_wmma_kernel<256, 128, true>
        <<<MB8 * (128 / 16), T, 0, stream>>>(agg, inv_deg, h1, Wl2, Wr2, bl2, h2);

    // --- layer 3: 128 -> 128, no relu -------------------------------------
    fill_zero4_kernel<<<ZB, T, 0, stream>>>((float4*)agg, (long long)N_NODES * 128 / 4);
    scatter_add4_kernel<<<SB, T, 0, stream>>>(h2, src, dst, agg, N_EDGES, 128 / 4);
    sage_gemm_wmma_kernel<128, 128, false>
        <<<MB8 * (128 / 16), T, 0, stream>>>(agg, inv_deg, h2, Wl3, Wr3, bl3, outf);
}